// CRF_62234076119650
// MI455X (gfx1250) — compile-verified
//
#include <hip/hip_runtime.h>
#include <hip/hip_bf16.h>

typedef __attribute__((ext_vector_type(2))) float v2f;
typedef __attribute__((ext_vector_type(8))) float v8f;

#define T_DIM 1024
#define L_DIM 32
#define K_DIM 128
#define N_ROWS (T_DIM * L_DIM * L_DIM)   // 1,048,576 rows of length 128
#define N_TILES (N_ROWS / 16)            // 65,536 tiles of 16 rows

// ---------------------------------------------------------------------------
// Kernel 1: M[r] = dot(X[r, 0:128], w)  via V_WMMA_F32_16X16X4_F32.
// A-matrix (16x4) = w chunk replicated across all 16 rows (precomputed in
// regs; documented layout: v0 = K0 (lanes 0-15) / K2 (lanes 16-31), v1 = K1/K3).
// B-matrix (4x16) = 16 X-rows; lane n supplies column n: a float2 of row
// (tile*16 + (lane&15)) at k-offset (4*kc + (lane>=16 ? 2 : 0)).
// D[0, n] = dot(X_row_n, w) lands in acc[0] across lanes 0-15.
// Bandwidth-bound: 512 MB of X -> ~22 us floor at 23.3 TB/s.
// ---------------------------------------------------------------------------
__global__ __launch_bounds__(256) void crf_gemv_wmma(
    const float* __restrict__ X, const float* __restrict__ w,
    float* __restrict__ M) {
  const int lane   = threadIdx.x & 31;
  const int half   = lane >> 4;          // 0: lanes 0-15, 1: lanes 16-31
  const int mrow   = lane & 15;
  const int wavesPerBlock = blockDim.x >> 5;
  const int gwave  = blockIdx.x * wavesPerBlock + (threadIdx.x >> 5);
  const int nwaves = gridDim.x * wavesPerBlock;

  // Pre-build all 32 A-chunks of w (stays in 64 VGPRs for the whole kernel).
  v2f Aw[32];
#pragma unroll
  for (int kc = 0; kc < 32; ++kc) {
    const int k0 = kc * 4 + half * 2;
    Aw[kc].x = w[k0];
    Aw[kc].y = w[k0 + 1];
  }

  for (int tile = gwave; tile < N_TILES; tile += nwaves) {
    const float* rowp =
        X + ((size_t)tile * 16 + (size_t)mrow) * K_DIM + half * 2;
    v8f acc = {};
#pragma unroll
    for (int kc = 0; kc < 32; ++kc) {
      v2f b = *(const v2f*)(rowp + kc * 4);  // 8B-aligned float2 of this row
      // 8 args: (neg_a, A, neg_b, B, c_mod, C, reuse_a, reuse_b)
      acc = __builtin_amdgcn_wmma_f32_16x16x4_f32(
          false, Aw[kc], false, b, (short)0, acc, false, false);
    }
    if (lane < 16) M[tile * 16 + mrow] = acc[0];
  }
}

// ---------------------------------------------------------------------------
// Kernel 2: per-timestep rescale. Block t: c[t] = max_{ij} M[t], then
// P[t,i,j] = exp(M[t,i,j] - c[t]).  8 MB total traffic, lives in L2.
// ---------------------------------------------------------------------------
__global__ __launch_bounds__(256) void crf_max_exp(
    const float* __restrict__ M, float* __restrict__ P,
    float* __restrict__ c) {
  __shared__ float red[256];
  const int t   = blockIdx.x;
  const int tid = threadIdx.x;
  const float* Mt = M + (size_t)t * 1024;
  float v[4];
  float m = -__builtin_inff();
#pragma unroll
  for (int r = 0; r < 4; ++r) {
    v[r] = Mt[tid + r * 256];
    m = fmaxf(m, v[r]);
  }
  red[tid] = m;
  __syncthreads();
  for (int s = 128; s > 0; s >>= 1) {
    if (tid < s) red[tid] = fmaxf(red[tid], red[tid + s]);
    __syncthreads();
  }
  const float ct = red[0];
#pragma unroll
  for (int r = 0; r < 4; ++r)
    P[(size_t)t * 1024 + tid + r * 256] = __expf(v[r] - ct);
  if (tid == 0) c[t] = ct;
}

// ---------------------------------------------------------------------------
// Kernel 3: single wave32, lane j <-> CRF state j.
// path_score gather + exp-domain forward recursion + final logsumexp.
// Per step: wave-max(alpha), e = exp(alpha - m), y_j = sum_i e_i * P[t,i,j]
// via v_readlane broadcasts (scalar FMA operand) and 4 FMA chains, then
// alpha_j = m + c_t + log(y_j).  Exact same math as the reference up to
// rounding (the per-step shift cancels inside logsumexp).
// ---------------------------------------------------------------------------
__device__ __forceinline__ float readlane_f(float v, int l) {
  return __builtin_bit_cast(
      float, __builtin_amdgcn_readlane(__builtin_bit_cast(int, v), l));
}

__global__ void crf_scan(const float* __restrict__ M,
                         const float* __restrict__ P,
                         const float* __restrict__ c,
                         const int* __restrict__ label,
                         float* __restrict__ out) {
  const int lane = threadIdx.x;  // 0..31, one wave

  // ---- path score: sum_t M[t, prev_t, label_t], BEG = 0 ----
  float ps = 0.0f;
  for (int t = lane; t < T_DIM; t += 32) {
    const int prev = (t == 0) ? 0 : label[t - 1];
    ps += M[(size_t)t * 1024 + prev * 32 + label[t]];
  }
#pragma unroll
  for (int s = 16; s > 0; s >>= 1) ps += __shfl_xor(ps, s, 32);

  // ---- forward recursion ----
  float alpha = M[lane];  // Ms[0, BEG=0, :]
  for (int t = 1; t < T_DIM; ++t) {
    float m = alpha;
#pragma unroll
    for (int s = 16; s > 0; s >>= 1) m = fmaxf(m, __shfl_xor(m, s, 32));
    const float e = __expf(alpha - m);
    const float* Pt = P + (size_t)t * 1024 + lane;  // column j = lane
    float y0 = 0.f, y1 = 0.f, y2 = 0.f, y3 = 0.f;
#pragma unroll
    for (int i = 0; i < 32; i += 4) {
      const float p0 = Pt[(i + 0) * 32];
      const float p1 = Pt[(i + 1) * 32];
      const float p2 = Pt[(i + 2) * 32];
      const float p3 = Pt[(i + 3) * 32];
      y0 = fmaf(readlane_f(e, i + 0), p0, y0);
      y1 = fmaf(readlane_f(e, i + 1), p1, y1);
      y2 = fmaf(readlane_f(e, i + 2), p2, y2);
      y3 = fmaf(readlane_f(e, i + 3), p3, y3);
    }
    const float y = (y0 + y1) + (y2 + y3);
    alpha = m + c[t] + __logf(y);
  }

  // ---- logZ = logsumexp(alpha) ----
  float m2 = alpha;
#pragma unroll
  for (int s = 16; s > 0; s >>= 1) m2 = fmaxf(m2, __shfl_xor(m2, s, 32));
  float s2 = __expf(alpha - m2);
#pragma unroll
  for (int s = 16; s > 0; s >>= 1) s2 += __shfl_xor(s2, s, 32);

  if (lane == 0) out[0] = -ps + (m2 + __logf(s2));
}

// ---------------------------------------------------------------------------
extern "C" void kernel_launch(void* const* d_in, const int* in_sizes, int n_in,
                              void* d_out, int out_size, void* d_ws,
                              size_t ws_size, hipStream_t stream) {
  const float* X     = (const float*)d_in[0];
  const float* w     = (const float*)d_in[1];
  const int*   label = (const int*)d_in[2];
  float* out = (float*)d_out;

  // Workspace layout: M (4 MB) | P (4 MB) | c (4 KB)  => ~8.4 MB total.
  float* M = (float*)d_ws;
  float* P = M + (size_t)N_ROWS;
  float* c = P + (size_t)N_ROWS;

  crf_gemv_wmma<<<2048, 256, 0, stream>>>(X, w, M);   // 16384 waves, 4 tiles each
  crf_max_exp<<<T_DIM, 256, 0, stream>>>(M, P, c);
  crf_scan<<<1, 32, 0, stream>>>(M, P, c, label, out);
}